// SSIMLoss_57071525429693
// MI455X (gfx1250) — compile-verified
//
#include <hip/hip_runtime.h>

typedef float v2f __attribute__((ext_vector_type(2)));
typedef float v8f __attribute__((ext_vector_type(8)));

#define IMG      512
#define TILE     16
#define PADR     5          // (11-1)/2
#define HROWS    26         // TILE + 2*PADR (valid input rows per tile)
#define HROWSA   32         // allocated H rows (2 groups of 16)
#define HPITCH   18         // LDS row pitch (floats): conflict-free for stores & gathers
#define WAVES    4          // waves per block
#define C1F      1.0e-4f    // (0.01)^2
#define C2F      9.0e-4f    // (0.03)^2
#define NPIX_INV (1.0f / 16777216.0f)   // 1/(64*512*512)

__global__ void ssim_zero_kernel(float* acc) {
    if (threadIdx.x == 0 && blockIdx.x == 0) acc[0] = 0.0f;
}

__global__ void ssim_finalize_kernel(const float* acc, float* out) {
    if (threadIdx.x == 0 && blockIdx.x == 0)
        out[0] = 1.0f - acc[0] * NPIX_INV;
}

__global__ __launch_bounds__(WAVES * 32)
void ssim_main_kernel(const float* __restrict__ pred,
                      const float* __restrict__ targ,
                      float* __restrict__ acc_out) {
    // 5 horizontally-convolved fields per wave: p, t, p*p, t*t, p*t
    __shared__ float Hs[WAVES][5][HROWSA][HPITCH];

    const int wave = threadIdx.x >> 5;
    const int lane = threadIdx.x & 31;

    // one wave == one 16x16 output tile; 64 batches * 32 * 32 tiles = 65536
    const int tileId = blockIdx.x * WAVES + wave;
    const int b  = tileId >> 10;
    const int ty = (tileId >> 5) & 31;
    const int tx = tileId & 31;
    const int x0 = tx * TILE;
    const int y0 = ty * TILE;

    const float* __restrict__ pb = pred + (size_t)b * (IMG * IMG);
    const float* __restrict__ tb = targ + (size_t)b * (IMG * IMG);

    // Per-lane fragment indices shared by every WMMA in this kernel:
    //   16-side index  = lane & 15   (M for A-operands, N for B-operands)
    //   K index        = 4*t + v + 2*(lane>=16)
    const int m  = lane & 15;
    const int j0 = (lane >> 4) << 1;

    // ---- banded Gaussian fragments: band[j][i16] = g[j - i16], 0 <= j-i16 <= 10 ----
    // Used as B in the horizontal pass and as A in the vertical pass (identical
    // per-lane values in both roles).  g[d] = exp(-(d-5)^2/4.5)/3.759232
    v2f wb[7];
#pragma unroll
    for (int tk = 0; tk < 7; ++tk) {
#pragma unroll
        for (int v = 0; v < 2; ++v) {
            const int j  = 4 * tk + j0 + v;
            const int d  = j - m;
            const int dc = d - 5;
            const float g = __expf(-(float)(dc * dc) * (1.0f / 4.5f)) * 0.26601133f;
            wb[tk][v] = ((unsigned)d <= 10u) ? g : 0.0f;
        }
    }

    // ---- stage 1: horizontal pass as WMMA, data in A, banded weights in B ----
    // H[r][n] = sum_j in[r][x0-5+j] * band[j][n]; two row-groups of 16.
    // Rows >= 26 are never weighted in the vertical band, so only loads are guarded.
#pragma unroll
    for (int g = 0; g < 2; ++g) {
        const int rg = 16 * g + m;              // H row this lane supplies (A's M)
        const int gy = y0 - PADR + rg;
        const bool vrow = (rg < HROWS) && (gy >= 0) && (gy < IMG);

        float pa[14], ta[14];
#pragma unroll
        for (int tk = 0; tk < 7; ++tk) {
#pragma unroll
            for (int v = 0; v < 2; ++v) {
                const int e  = 4 * tk + j0 + v;
                const int gx = x0 - PADR + e;
                const bool ok = vrow && (gx >= 0) && (gx < IMG);
                const int idx = gy * IMG + gx;
                pa[2 * tk + v] = ok ? pb[idx] : 0.0f;
                ta[2 * tk + v] = ok ? tb[idx] : 0.0f;
            }
        }

        v8f h0 = {}, h1 = {}, h2 = {}, h3 = {}, h4 = {};
#pragma unroll
        for (int tk = 0; tk < 7; ++tk) {
            v2f ap, at;
            ap.x = pa[2 * tk]; ap.y = pa[2 * tk + 1];
            at.x = ta[2 * tk]; at.y = ta[2 * tk + 1];
            const v2f app = ap * ap;            // elementwise -> v_pk_mul_f32
            const v2f att = at * at;
            const v2f apt = ap * at;
            h0 = __builtin_amdgcn_wmma_f32_16x16x4_f32(false, ap,  false, wb[tk], (short)0, h0, false, false);
            h1 = __builtin_amdgcn_wmma_f32_16x16x4_f32(false, at,  false, wb[tk], (short)0, h1, false, false);
            h2 = __builtin_amdgcn_wmma_f32_16x16x4_f32(false, app, false, wb[tk], (short)0, h2, false, false);
            h3 = __builtin_amdgcn_wmma_f32_16x16x4_f32(false, att, false, wb[tk], (short)0, h3, false, false);
            h4 = __builtin_amdgcn_wmma_f32_16x16x4_f32(false, apt, false, wb[tk], (short)0, h4, false, false);
        }

        // D layout: VGPR r -> row (16g + r) for lanes 0-15, (16g + r + 8) for lanes 16-31
#pragma unroll
        for (int r = 0; r < 8; ++r) {
            const int row = 16 * g + r + ((lane >> 4) << 3);
            const int o   = row * HPITCH + m;
            Hs[wave][0][0][o] = h0[r];
            Hs[wave][1][0][o] = h1[r];
            Hs[wave][2][0][o] = h2[r];
            Hs[wave][3][0][o] = h3[r];
            Hs[wave][4][0][o] = h4[r];
        }
    }

    // intra-wave producer->consumer through LDS: make sure stores landed
    asm volatile("s_wait_dscnt 0x0" ::: "memory");

    // ---- stage 2: vertical pass, banded weights in A, H rows in B ----
    v8f acc[5];
#pragma unroll
    for (int f = 0; f < 5; ++f) {
        const float* __restrict__ Hf = &Hs[wave][f][0][0];
        v8f c = {};
#pragma unroll
        for (int tk = 0; tk < 7; ++tk) {
            const int j = 4 * tk + j0;
            v2f bfrag;
            bfrag.x = Hf[j * HPITCH + m];
            bfrag.y = Hf[(j + 1) * HPITCH + m];
            c = __builtin_amdgcn_wmma_f32_16x16x4_f32(
                    false, wb[tk], false, bfrag, (short)0, c, false, false);
        }
        acc[f] = c;
    }

    // ---- stage 3: SSIM map + reduction ----
    float ssum = 0.0f;
#pragma unroll
    for (int e = 0; e < 8; ++e) {
        const float mu1 = acc[0][e];
        const float mu2 = acc[1][e];
        const float mu1s = mu1 * mu1;
        const float mu2s = mu2 * mu2;
        const float mu12 = mu1 * mu2;
        const float s1  = acc[2][e] - mu1s;
        const float s2  = acc[3][e] - mu2s;
        const float s12 = acc[4][e] - mu12;
        const float num = (2.0f * mu12 + C1F) * (2.0f * s12 + C2F);
        const float den = (mu1s + mu2s + C1F) * (s1 + s2 + C2F);
        ssum += num / den;
    }

    // wave32 reduction
#pragma unroll
    for (int off = 16; off >= 1; off >>= 1)
        ssum += __shfl_xor(ssum, off, 32);

    if (lane == 0) atomicAdd(acc_out, ssum);
}

extern "C" void kernel_launch(void* const* d_in, const int* in_sizes, int n_in,
                              void* d_out, int out_size, void* d_ws, size_t ws_size,
                              hipStream_t stream) {
    (void)in_sizes; (void)n_in; (void)out_size;
    const float* pred = (const float*)d_in[0];
    const float* targ = (const float*)d_in[1];
    float* out = (float*)d_out;
    float* acc = (ws_size >= sizeof(float)) ? (float*)d_ws : out;

    // 65536 tiles, 4 waves (tiles) per 128-thread block
    const int nblocks = 65536 / WAVES;

    ssim_zero_kernel<<<1, 1, 0, stream>>>(acc);
    ssim_main_kernel<<<nblocks, WAVES * 32, 0, stream>>>(pred, targ, acc);
    ssim_finalize_kernel<<<1, 1, 0, stream>>>(acc, out);
}